// RGLRU_80307298500854
// MI455X (gfx1250) — compile-verified
//
#include <hip/hip_runtime.h>
#include <hip/hip_bf16.h>
#include <math.h>

// RG-LRU (Griffin) forward for MI455X / gfx1250 (wave32, WMMA).
//
// Shapes: S=256, D_MODEL=512, D_RNN=512.
//   K1: z = sigmoid(x @ W1^T + b1)          -> fp32 WMMA 16x16x4, 1024 tiles
//   K2: a = exp(-8*softplus(Lambda)*rec), g = sqrt(1-a^2)*inp
//   K3: h = a_t*h + g_t*x_d scan over S; y[s,r,d] streamed with NT stores.
//
// Memory-bound: 269 MB of output -> ~11.6 us floor at 23.3 TB/s.

#define S_LEN   256
#define DMODEL  512
#define DRNN    512
#define NOUT    1024   // 2*DRNN

typedef __attribute__((ext_vector_type(2))) float v2f;
typedef __attribute__((ext_vector_type(8))) float v8f;

// ---------------------------------------------------------------------------
// K1: z[s, n] = sigmoid( sum_k x[s,k] * W1[n,k] + b1[n] )
// One 16x16 output tile per wave. 8 waves/block, 128 blocks = 1024 tiles.
// A-frag (f32 16x4): lane l -> row m0+(l&15), cols k+2*(l>>4) .. +1  (float2)
// B-frag (f32 4x16): lane l -> col n0+(l&15), rows k+2*(l>>4) .. +1  (float2,
//   B[k][n] = W1[n][k], so it's the same contiguous float2 load from W1).
// ---------------------------------------------------------------------------
__global__ void __launch_bounds__(256) rglru_gemm_sig(
    const float* __restrict__ x,    // (256, 512)
    const float* __restrict__ W1,   // (1024, 512)
    const float* __restrict__ b1,   // (1024,)
    float* __restrict__ z)          // (256, 1024)
{
  const int lane = threadIdx.x & 31;
  const int wave = threadIdx.x >> 5;
  const int tile = blockIdx.x * 8 + wave;      // 0..1023
  const int mt   = tile & 15;                  // 16 tiles along S
  const int nt   = tile >> 4;                  // 64 tiles along N
  const int m0   = mt * 16;
  const int n0   = nt * 16;
  const int lm   = lane & 15;
  const int half = lane >> 4;

  const float* arow = x  + (m0 + lm) * DMODEL + 2 * half;
  const float* brow = W1 + (n0 + lm) * DMODEL + 2 * half;

  v8f acc = {};
  for (int k = 0; k < DMODEL; k += 4) {
    v2f afrag = *(const v2f*)(arow + k);
    v2f bfrag = *(const v2f*)(brow + k);
    // 8 args: (neg_a, A, neg_b, B, c_mod, C, reuse_a, reuse_b)
    acc = __builtin_amdgcn_wmma_f32_16x16x4_f32(
        false, afrag, false, bfrag, (short)0, acc, false, false);
  }

  const float bias = b1[n0 + lm];
#pragma unroll
  for (int j = 0; j < 8; ++j) {
    const int m = m0 + half * 8 + j;           // C/D layout: vgpr j, lane half
    const float v = acc[j] + bias;
    z[m * NOUT + (n0 + lm)] = 1.0f / (1.0f + expf(-v));
  }
}

// ---------------------------------------------------------------------------
// K2: per (s, r): a = exp(-8 * softplus(Lambda[r]) * rec), g = sqrt(1-a^2)*inp
// ---------------------------------------------------------------------------
__global__ void __launch_bounds__(256) rglru_gates(
    const float* __restrict__ z,       // (256, 1024)
    const float* __restrict__ Lambda,  // (512,)
    float* __restrict__ a,             // (256, 512)
    float* __restrict__ g)             // (256, 512)
{
  const int idx = blockIdx.x * 256 + threadIdx.x;  // 0..131071
  const int s = idx >> 9;
  const int r = idx & (DRNN - 1);

  const float L  = Lambda[r];
  // numerically-stable softplus
  const float sp = log1pf(expf(-fabsf(L))) + fmaxf(L, 0.0f);

  const float inp = z[s * NOUT + r];
  const float rec = z[s * NOUT + DRNN + r];

  const float av = expf(-8.0f * sp * rec);
  const float gv = sqrtf(fmaxf(1.0f - av * av, 0.0f)) * inp;

  a[idx] = av;
  g[idx] = gv;
}

// ---------------------------------------------------------------------------
// K3: scan. Block = one r (512 blocks), 256 threads each own d and d+256.
// Per step: uniform scalar loads of a,g; coalesced x loads (cache-resident);
// 2 KB contiguous NT stores of y per block per step (y >> L2, stream it).
// ---------------------------------------------------------------------------
__global__ void __launch_bounds__(256) rglru_scan(
    const float* __restrict__ x,      // (256, 512)
    const float* __restrict__ a,      // (256, 512)
    const float* __restrict__ g,      // (256, 512)
    const float* __restrict__ h_in,   // (512, 512) initial state
    float* __restrict__ y,            // (256, 512, 512)
    float* __restrict__ h_out)        // (512, 512)
{
  const int r  = blockIdx.x;          // 0..511
  const int d0 = threadIdx.x;         // d0 and d0+256

  float h0 = h_in[(size_t)r * DMODEL + d0];
  float h1 = h_in[(size_t)r * DMODEL + d0 + 256];

  for (int s = 0; s < S_LEN; ++s) {
    // stay ahead on x (global_prefetch_b8); x is small & cache-resident
    if (s + 8 < S_LEN) {
      __builtin_prefetch(x + (s + 8) * DMODEL + d0, 0, 3);
    }
    const float av = a[s * DRNN + r];          // wave-uniform -> s_load
    const float gv = g[s * DRNN + r];          // wave-uniform -> s_load
    const float xa = x[s * DMODEL + d0];
    const float xb = x[s * DMODEL + d0 + 256];

    h0 = fmaf(av, h0, gv * xa);
    h1 = fmaf(av, h1, gv * xb);

    float* yr = y + ((size_t)s * DRNN + r) * DMODEL;
    __builtin_nontemporal_store(h0, yr + d0);        // streaming, bypass L2 rinse
    __builtin_nontemporal_store(h1, yr + d0 + 256);
  }

  __builtin_nontemporal_store(h0, h_out + (size_t)r * DMODEL + d0);
  __builtin_nontemporal_store(h1, h_out + (size_t)r * DMODEL + d0 + 256);
}

// ---------------------------------------------------------------------------
extern "C" void kernel_launch(void* const* d_in, const int* in_sizes, int n_in,
                              void* d_out, int out_size, void* d_ws, size_t ws_size,
                              hipStream_t stream) {
  const float* x      = (const float*)d_in[0];   // (256, 512)
  const float* state0 = (const float*)d_in[1];   // (512, 512)
  const float* W1     = (const float*)d_in[2];   // (1024, 512)
  const float* b1     = (const float*)d_in[3];   // (1024,)
  const float* Lambda = (const float*)d_in[4];   // (512,)

  float* y     = (float*)d_out;                          // 256*512*512
  float* state = y + (size_t)S_LEN * DRNN * DMODEL;      // (512, 512)

  // workspace: z (256x1024), a (256x512), g (256x512) -> 2 MB total
  float* z = (float*)d_ws;
  float* a = z + S_LEN * NOUT;
  float* g = a + S_LEN * DRNN;

  rglru_gemm_sig<<<128, 256, 0, stream>>>(x, W1, b1, z);
  rglru_gates  <<<512, 256, 0, stream>>>(z, Lambda, a, g);
  rglru_scan   <<<512, 256, 0, stream>>>(x, a, g, state0, y, state);
}